// QCCPNLoss_65412351918314
// MI455X (gfx1250) — compile-verified
//
#include <hip/hip_runtime.h>

typedef float v2f __attribute__((ext_vector_type(2)));
typedef float v8f __attribute__((ext_vector_type(8)));

#define GAMMA_ 0.1f
#define EPSC_  20.0f
#define DDIM   128
#define BSTRIDE 132                 // padded LDS row stride (conflict-free)
#define BTILE   (16 * BSTRIDE)      // floats per staged B tile

// ---------------- Kernel 1: center squared norms (+ zero the output scalar) ----
__global__ void k_center_norms(const float* __restrict__ centers,
                               float* __restrict__ cn,
                               float* __restrict__ out, int C) {
  __shared__ float red[128];
  const int j = blockIdx.x;
  if (j == 0 && threadIdx.x == 0) out[0] = 0.0f;   // runs before main kernel (stream order)
  float v = centers[(size_t)j * DDIM + threadIdx.x];
  red[threadIdx.x] = v * v;
  __syncthreads();
  for (int s = 64; s > 0; s >>= 1) {
    if (threadIdx.x < s) red[threadIdx.x] += red[threadIdx.x + s];
    __syncthreads();
  }
  if (threadIdx.x == 0) cn[j] = red[0];
}

// ---------------- Kernel 2: per-center mean distance to all OTHER centers -----
__global__ void k_center_contrast(const float* __restrict__ centers,
                                  const float* __restrict__ cn,
                                  float* __restrict__ rowmean, int C) {
  __shared__ float cj[DDIM];
  __shared__ float red[128];
  const int j = blockIdx.x;
  cj[threadIdx.x] = centers[(size_t)j * DDIM + threadIdx.x];
  __syncthreads();
  const float cnj = cn[j];
  float sum = 0.f;
  for (int j2 = threadIdx.x; j2 < C; j2 += 128) {
    const float* c2 = centers + (size_t)j2 * DDIM;
    float dot = 0.f;
#pragma unroll 8
    for (int k = 0; k < DDIM; k++) dot = fmaf(cj[k], c2[k], dot);
    if (j2 != j) {
      float sq = cnj + cn[j2] - 2.f * dot;
      sum += sqrtf(fmaxf(sq, 0.f));
    }
  }
  red[threadIdx.x] = sum;
  __syncthreads();
  for (int s = 64; s > 0; s >>= 1) {
    if (threadIdx.x < s) red[threadIdx.x] += red[threadIdx.x + s];
    __syncthreads();
  }
  if (threadIdx.x == 0) rowmean[j] = red[0] / (float)(C - 1);
}

// Issue the async global->LDS copy of one 16x128 B tile (padded rows in LDS).
// 512 b128 chunks; 256 threads issue 2 each. Rows beyond C clamped (masked later).
__device__ __forceinline__ void prefetch_tile(const float* __restrict__ centers,
                                              int jt, int C, float* buf) {
  const unsigned lds0 = (unsigned)(uintptr_t)buf;   // generic ptr low bits = LDS offset
  const int t = threadIdx.x;
#pragma unroll
  for (int i = 0; i < 2; i++) {
    const int c    = t + i * 256;     // chunk id 0..511
    const int row  = c >> 5;          // 0..15
    const int colq = c & 31;          // 0..31 (4 floats each)
    int gr = jt * 16 + row;
    if (gr >= C) gr = 0;
    const float* gp = centers + (size_t)gr * DDIM + colq * 4;
    const unsigned lds = lds0 + (unsigned)((row * BSTRIDE + colq * 4) * 4);
    asm volatile("global_load_async_to_lds_b128 %0, %1, off"
                 :: "v"(lds), "v"(gp) : "memory");
  }
}

// ---------------- Kernel 3: fused WMMA-f32 GEMM + online log-softmax ----------
// A (16x128 per wave) register-resident; B tiles double-buffered in LDS via
// async global->LDS copies (ASYNCcnt), consumed as ds_load_b64 fragments.
__global__ __launch_bounds__(256) void k_dce_main(
    const float* __restrict__ features, const int* __restrict__ labels,
    const float* __restrict__ centers, const float* __restrict__ cn,
    const float* __restrict__ rowmean, float* __restrict__ out,
    int Bn, int C) {
  __shared__ float btile[2][BTILE];

  const int lane = threadIdx.x & 31;
  const int wave = threadIdx.x >> 5;
  const int half = lane >> 4;   // which 16-lane half of the wave
  const int l16  = lane & 15;
  const int rowBase = (blockIdx.x * 8 + wave) * 16;

  // --- A fragments: 32-bit A 16x4 layout: M = lane&15 on both halves,
  //     lanes 0-15 carry K = {4k, 4k+1}, lanes 16-31 carry K = {4k+2, 4k+3}.
  int rA = rowBase + l16; if (rA >= Bn) rA = Bn - 1;
  const float* fr = features + (size_t)rA * DDIM;
  v2f a[32];
  float pn = 0.f;                               // partial ||f_row||^2
#pragma unroll
  for (int kc = 0; kc < 32; kc++) {
    const int k = kc * 4 + 2 * half;
    v2f t; t.x = fr[k]; t.y = fr[k + 1];
    a[kc] = t;
    pn = fmaf(t.x, t.x, fmaf(t.y, t.y, pn));
  }
  // row norms for the 8 rows this half holds in the C/D tile (M = v + 8*half)
  float fn[8];
#pragma unroll
  for (int v = 0; v < 8; v++) {
    int src = v + 8 * half;
    fn[v] = __shfl(pn, src, 32) + __shfl(pn, src + 16, 32);
  }
  int lab[8];
#pragma unroll
  for (int v = 0; v < 8; v++) {
    int r = rowBase + v + 8 * half;
    lab[v] = labels[r < Bn ? r : (Bn - 1)];
  }

  float mrun[8], srun[8], dlab[8];
#pragma unroll
  for (int v = 0; v < 8; v++) { mrun[v] = -3.0e38f; srun[v] = 0.f; dlab[v] = 0.f; }

  const int ntiles = (C + 15) >> 4;

  // prologue: stage tile 0
  prefetch_tile(centers, 0, C, &btile[0][0]);
  asm volatile("s_wait_asynccnt 0" ::: "memory");
  __syncthreads();

  for (int jt = 0; jt < ntiles; jt++) {
    const int cur = jt & 1;
    if (jt + 1 < ntiles)                          // overlap copy of jt+1 with compute of jt
      prefetch_tile(centers, jt + 1, C, &btile[1 - cur][0]);

    const int j = jt * 16 + l16;          // this lane's output column (N = lane&15)
    const bool valid = (j < C);
    const int js = valid ? j : 0;
    v8f acc = {0.f, 0.f, 0.f, 0.f, 0.f, 0.f, 0.f, 0.f};
#pragma unroll
    for (int kc = 0; kc < 32; kc++) {
      const int k = kc * 4 + 2 * half;    // B 4x16: N on lanes, K split like A
      v2f b;
      b.x = btile[cur][l16 * BSTRIDE + k];
      b.y = btile[cur][l16 * BSTRIDE + k + 1];
      acc = __builtin_amdgcn_wmma_f32_16x16x4_f32(
          false, a[kc], false, b, (short)0, acc, false, false);
    }
    const float cnj = cn[js];
#pragma unroll
    for (int v = 0; v < 8; v++) {
      float sq    = fn[v] - 2.f * acc[v] + cnj;
      float dist  = sqrtf(fmaxf(sq, 0.f));
      float logit = -GAMMA_ * dist;
      if (valid) {
        float mn = fmaxf(mrun[v], logit);
        srun[v]  = srun[v] * __expf(mrun[v] - mn) + __expf(logit - mn);
        mrun[v]  = mn;
        if (j == lab[v]) dlab[v] += GAMMA_ * dist;
      }
    }

    // next buffer must be fully written (our own async ops) before all waves use it
    asm volatile("s_wait_asynccnt 0" ::: "memory");
    __syncthreads();
  }

  // merge the 16 lanes of each half (xor masks 1,2,4,8 stay within the half)
#pragma unroll
  for (int off = 1; off < 16; off <<= 1) {
#pragma unroll
    for (int v = 0; v < 8; v++) {
      float m2 = __shfl_xor(mrun[v], off, 32);
      float s2 = __shfl_xor(srun[v], off, 32);
      float mn = fmaxf(mrun[v], m2);
      srun[v] = srun[v] * __expf(mrun[v] - mn) + s2 * __expf(m2 - mn);
      mrun[v] = mn;
      dlab[v] += __shfl_xor(dlab[v], off, 32);
    }
  }

  if (l16 == 0) {                         // lanes 0 and 16 each own 8 rows
    float accsum = 0.f;
#pragma unroll
    for (int v = 0; v < 8; v++) {
      int r = rowBase + v + 8 * half;
      if (r < Bn) {
        float lse = mrun[v] + __logf(srun[v]);                 // log-sum-exp
        float h   = fmaxf(EPSC_ - rowmean[lab[v]], 0.f);       // hinge term
        accsum += lse + dlab[v] + h;                           // +gamma*dist[label]
      }
    }
    atomicAdd(out, accsum / (float)Bn);
  }
}

// ------------------------------- launcher -------------------------------------
extern "C" void kernel_launch(void* const* d_in, const int* in_sizes, int n_in,
                              void* d_out, int out_size, void* d_ws, size_t ws_size,
                              hipStream_t stream) {
  const float* features = (const float*)d_in[0];
  const int*   labels   = (const int*)d_in[1];
  const float* centers  = (const float*)d_in[2];
  float* out = (float*)d_out;

  const int Bn = in_sizes[1];            // 16384
  const int C  = in_sizes[2] / DDIM;     // 1000

  float* cn      = (float*)d_ws;                    // [C] center sq-norms
  float* rowmean = cn + ((C + 255) & ~255);         // [C] mean off-diag dist

  k_center_norms<<<C, 128, 0, stream>>>(centers, cn, out, C);
  k_center_contrast<<<C, 128, 0, stream>>>(centers, cn, rowmean, C);

  const int nblocks = (Bn + 127) / 128;  // 8 waves/block, 16 rows/wave
  k_dce_main<<<nblocks, 256, 0, stream>>>(features, labels, centers,
                                          cn, rowmean, out, Bn, C);
}